// TripletLoss_15530601742671
// MI455X (gfx1250) — compile-verified
//
#include <hip/hip_runtime.h>
#include <hip/hip_bf16.h>

#define NN 384
#define DD 512
#define NTILES 24   // 384 / 16

typedef float v2f __attribute__((ext_vector_type(2)));
typedef float v4f __attribute__((ext_vector_type(4)));
typedef float v8f __attribute__((ext_vector_type(8)));

// ---------------------------------------------------------------------------
// Kernel 1: Gram matrix G = X * X^T via V_WMMA_F32_16X16X4_F32 (full fp32).
// One wave (32 lanes) per 16x16 output tile; 24x24 = 576 tiles = 72 blocks x 8 waves.
//
// A (16x4 f32) lane layout: lane L, VGPR v -> M = L%16, K = (L>>4)*2 + v
// B (4x16 f32) lane layout: lane L, VGPR v -> N = L%16, K = (L>>4)*2 + v
//   and B[k][n] = X[n][k], so A-loads and B-loads have identical addressing,
//   just from different row blocks.
// C/D (16x16 f32): lane L, VGPR v -> M = v + (L>>4)*8, N = L%16.
// We process K in chunks of 8 with one float4 load per matrix per chunk and
// two WMMAs; the induced K-permutation is identical for A and B, so the dot
// product over K=512 is exact.
// ---------------------------------------------------------------------------
__global__ __launch_bounds__(256) void gram_wmma_kernel(
    const float* __restrict__ X, float* __restrict__ G) {
  const int wave = (blockIdx.x * blockDim.x + threadIdx.x) >> 5;
  const int lane = threadIdx.x & 31;
  const int ti = wave / NTILES;      // row tile
  const int tj = wave % NTILES;      // col tile
  const int m  = lane & 15;
  const int hi = lane >> 4;

  const float* __restrict__ arow = X + (ti * 16 + m) * DD;
  const float* __restrict__ brow = X + (tj * 16 + m) * DD;

  v8f c = {0.f, 0.f, 0.f, 0.f, 0.f, 0.f, 0.f, 0.f};

  for (int k = 0; k < DD; k += 8) {
    v4f av = *(const v4f*)(arow + k + hi * 4);
    v4f bv = *(const v4f*)(brow + k + hi * 4);
    v2f a0 = {av.x, av.y};
    v2f b0 = {bv.x, bv.y};
    v2f a1 = {av.z, av.w};
    v2f b1 = {bv.z, bv.w};
    // 8 args: (neg_a, A, neg_b, B, c_mod, C, reuse_a, reuse_b)
    c = __builtin_amdgcn_wmma_f32_16x16x4_f32(false, a0, false, b0,
                                              (short)0, c, false, false);
    c = __builtin_amdgcn_wmma_f32_16x16x4_f32(false, a1, false, b1,
                                              (short)0, c, false, false);
  }

  const int col = tj * 16 + m;
#pragma unroll
  for (int v = 0; v < 8; ++v) {
    const int row = ti * 16 + v + hi * 8;
    G[row * NN + col] = c[v];
  }
}

// ---------------------------------------------------------------------------
// Kernel 2: extract squared norms from the Gram diagonal (sq[i] = G[i][i]).
// Must run before the in-place dist transform.
// ---------------------------------------------------------------------------
__global__ void diag_kernel(const float* __restrict__ G, float* __restrict__ sq) {
  int i = blockIdx.x * blockDim.x + threadIdx.x;
  if (i < NN) sq[i] = G[i * NN + i];
}

// ---------------------------------------------------------------------------
// Kernel 3: in-place G -> dist.  d2 = sq_i + sq_j - 2*G_ij, clamped; safe sqrt
// (diagonal is exactly 0 because sq comes from the same Gram values).
// ---------------------------------------------------------------------------
__global__ __launch_bounds__(256) void dist_kernel(
    float* __restrict__ G, const float* __restrict__ sq) {
  int idx = blockIdx.x * blockDim.x + threadIdx.x;
  if (idx >= NN * NN) return;
  int i = idx / NN;
  int j = idx - i * NN;
  float d2 = sq[i] + sq[j] - 2.0f * G[idx];
  d2 = fmaxf(d2, 0.0f);
  G[idx] = (d2 > 0.0f) ? sqrtf(d2) : 0.0f;
}

// ---------------------------------------------------------------------------
// Kernel 4: per-anchor batch-all triplet partial sums.
// loss_a = sum_{p: lab[p]==lab[a]} sum_{n: lab[n]!=lab[a]} relu(d_ap - d_an + 1)
// cnt_a  = |P_a| * |N_a|   (p==a included, matching the reference's pos mask)
// One block per anchor; the anchor's distance row + labels are staged in LDS.
// Deterministic: fixed-shape shared-memory tree reduction, no float atomics.
// ---------------------------------------------------------------------------
__global__ __launch_bounds__(256) void triplet_kernel(
    const float* __restrict__ dist, const int* __restrict__ labels,
    float* __restrict__ partial_loss, float* __restrict__ partial_cnt) {
  __shared__ float row[NN];
  __shared__ int   lab[NN];
  __shared__ float sL[256];
  __shared__ float sC[256];

  const int a   = blockIdx.x;
  const int tid = threadIdx.x;

  for (int j = tid; j < NN; j += 256) {
    row[j] = dist[a * NN + j];
    lab[j] = labels[j];
  }
  __syncthreads();

  const int la = lab[a];
  float loss = 0.0f;
  float cnt  = 0.0f;

  for (int p = tid; p < NN; p += 256) {
    if (lab[p] != la) continue;
    const float dp1 = row[p] + 1.0f;  // d_ap + margin
    float s = 0.0f, c = 0.0f;
    for (int n = 0; n < NN; ++n) {
      if (lab[n] == la) continue;
      s += fmaxf(dp1 - row[n], 0.0f);
      c += 1.0f;
    }
    loss += s;
    cnt  += c;
  }

  sL[tid] = loss;
  sC[tid] = cnt;
  __syncthreads();
#pragma unroll
  for (int o = 128; o > 0; o >>= 1) {
    if (tid < o) { sL[tid] += sL[tid + o]; sC[tid] += sC[tid + o]; }
    __syncthreads();
  }
  if (tid == 0) {
    partial_loss[a] = sL[0];
    partial_cnt[a]  = sC[0];
  }
}

// ---------------------------------------------------------------------------
// Kernel 5: final deterministic reduction over 384 anchors -> scalar.
// ---------------------------------------------------------------------------
__global__ __launch_bounds__(256) void final_kernel(
    const float* __restrict__ partial_loss, const float* __restrict__ partial_cnt,
    float* __restrict__ out) {
  __shared__ float sL[256];
  __shared__ float sC[256];
  const int tid = threadIdx.x;
  float l = 0.0f, c = 0.0f;
  for (int i = tid; i < NN; i += 256) { l += partial_loss[i]; c += partial_cnt[i]; }
  sL[tid] = l;
  sC[tid] = c;
  __syncthreads();
#pragma unroll
  for (int o = 128; o > 0; o >>= 1) {
    if (tid < o) { sL[tid] += sL[tid + o]; sC[tid] += sC[tid + o]; }
    __syncthreads();
  }
  if (tid == 0) out[0] = sL[0] / (sC[0] + 1e-16f);
}

// ---------------------------------------------------------------------------
extern "C" void kernel_launch(void* const* d_in, const int* in_sizes, int n_in,
                              void* d_out, int out_size, void* d_ws, size_t ws_size,
                              hipStream_t stream) {
  const float* X      = (const float*)d_in[0];  // [384, 512] fp32
  const int*   labels = (const int*)d_in[1];    // [384]
  float*       out    = (float*)d_out;          // scalar fp32

  // Workspace layout (~594 KB): G/dist (in place) | sq | partial loss | partial cnt
  float* G  = (float*)d_ws;          // NN*NN
  float* sq = G + NN * NN;           // NN
  float* pl = sq + NN;               // NN
  float* pc = pl + NN;               // NN

  // 576 tiles * 32 lanes = 18432 threads -> 72 blocks of 256 (8 waves/block)
  gram_wmma_kernel<<<(NTILES * NTILES * 32) / 256, 256, 0, stream>>>(X, G);
  diag_kernel<<<(NN + 255) / 256, 256, 0, stream>>>(G, sq);
  dist_kernel<<<(NN * NN + 255) / 256, 256, 0, stream>>>(G, sq);
  triplet_kernel<<<NN, 256, 0, stream>>>(G, labels, pl, pc);
  final_kernel<<<1, 256, 0, stream>>>(pl, pc, out);
}